// HybridAttention_55714315764274
// MI455X (gfx1250) — compile-verified
//
#include <hip/hip_runtime.h>
#include <hip/hip_bf16.h>
#include <math.h>

typedef __attribute__((ext_vector_type(2))) float v2f;
typedef __attribute__((ext_vector_type(8))) float v8f;

#define B_ 32
#define S_ 4096
#define H_ 512
#define NCH 32
#define SCHUNK (S_ / NCH) /* 128 */

// Workspace layout (in floats):
#define WS_TARGET   0                              // [2][32][512]   = 32768
#define WS_PM       32768                          // [2048]
#define WS_PS       34816                          // [2048]
#define WS_PW       36864                          // [2048][512]    = 1048576
#define WS_MZ       1085440                        // [64][2]
#define WS_WEIGHTED 1085568                        // [2][32][512]   = 32768
// total = 1,118,336 floats ~= 4.3 MB

// ---------------------------------------------------------------------------
// Kernel 1: target = inputs @ W_in for both branches, via V_WMMA_F32_16X16X4_F32
// grid: 2 branches * 2 Mtiles * 32 Ntiles = 128 blocks of 1 wave
// ---------------------------------------------------------------------------
__global__ __launch_bounds__(32) void target_gemm_kernel(
    const float* __restrict__ tree_in, const float* __restrict__ text_in,
    const float* __restrict__ W_in, float* __restrict__ ws)
{
    int bid = blockIdx.x;
    int br = bid >> 6;
    int mt = (bid >> 5) & 1;
    int nt = bid & 31;
    const float* A = br ? text_in : tree_in;

    int lane = threadIdx.x;
    int half = lane >> 4;       // 0: K={0,1}; 1: K={2,3}
    int l15  = lane & 15;
    int arow = mt * 16 + l15;   // b index
    int bcol = nt * 16 + l15;   // output column

    v8f acc = {0.f, 0.f, 0.f, 0.f, 0.f, 0.f, 0.f, 0.f};
    for (int k0 = 0; k0 < H_; k0 += 4) {
        int ka = k0 + half * 2;
        v2f a; a.x = A[arow * H_ + ka];        a.y = A[arow * H_ + ka + 1];
        v2f b; b.x = W_in[ka * H_ + bcol];     b.y = W_in[(ka + 1) * H_ + bcol];
        acc = __builtin_amdgcn_wmma_f32_16x16x4_f32(false, a, false, b,
                                                    (short)0, acc, false, false);
    }
    float* tgt = ws + WS_TARGET + (size_t)br * B_ * H_;
#pragma unroll
    for (int i = 0; i < 8; ++i) {
        int row = mt * 16 + i + half * 8;
        tgt[row * H_ + bcol] = acc[i];
    }
}

// ---------------------------------------------------------------------------
// Kernel 2: fused scores + online softmax partials + weighted partial sums.
// One wave per (branch, b, s-chunk): streams 128x512 f32 of context ONCE.
// grid: 2*32*32 = 2048 blocks of 32 threads.
// ---------------------------------------------------------------------------
__global__ __launch_bounds__(32) void attn_partial_kernel(
    const float* __restrict__ tree_ctx, const float* __restrict__ text_ctx,
    const float* __restrict__ tree_mask, const float* __restrict__ text_mask,
    float* __restrict__ ws, float* __restrict__ out)
{
    int bid = blockIdx.x;
    int br  = bid >> 10;
    int rem = bid & 1023;
    int b   = rem >> 5;
    int ch  = rem & 31;

    const float* ctx  = br ? text_ctx : tree_ctx;
    const float* mask = br ? text_mask : tree_mask;
    float* att = out + 16384 + (size_t)br * (B_ * S_) + (size_t)b * S_;

    int lane = threadIdx.x;
    const float* tgt = ws + WS_TARGET + ((size_t)br * B_ + b) * H_;

    float tg[16];
#pragma unroll
    for (int i = 0; i < 4; ++i) {
        float4 t4 = *(const float4*)(tgt + i * 128 + lane * 4);
        tg[i*4+0] = t4.x; tg[i*4+1] = t4.y; tg[i*4+2] = t4.z; tg[i*4+3] = t4.w;
    }

    float w[16];
#pragma unroll
    for (int i = 0; i < 16; ++i) w[i] = 0.f;
    float m = -INFINITY, ssum = 0.f;
    float keep = 0.f;

    int s0 = ch * SCHUNK;
    for (int s = s0; s < s0 + SCHUNK; ++s) {
        const float* row = ctx + ((size_t)b * S_ + s) * H_;
        float c[16];
#pragma unroll
        for (int i = 0; i < 4; ++i) {                // fully coalesced float4 loads
            float4 c4 = *(const float4*)(row + i * 128 + lane * 4);
            c[i*4+0] = c4.x; c[i*4+1] = c4.y; c[i*4+2] = c4.z; c[i*4+3] = c4.w;
        }
        float dp = 0.f;
#pragma unroll
        for (int i = 0; i < 16; ++i) dp = fmaf(c[i], tg[i], dp);
#pragma unroll
        for (int off = 16; off >= 1; off >>= 1)
            dp += __shfl_xor(dp, off, 32);

        float score = dp * mask[b * S_ + s];
        // stage raw masked scores, write coalesced 32 at a time
        if ((s & 31) == lane) keep = score;
        if ((s & 31) == 31) att[(s & ~31) + lane] = keep;

        // online softmax update with row kept in registers (single pass)
        if (score > m) {
            float sc = __expf(m - score);            // m=-inf first iter -> 0
            ssum = ssum * sc + 1.f;
#pragma unroll
            for (int i = 0; i < 16; ++i) w[i] = fmaf(w[i], sc, c[i]);
            m = score;
        } else {
            float p = __expf(score - m);
            ssum += p;
#pragma unroll
            for (int i = 0; i < 16; ++i) w[i] = fmaf(p, c[i], w[i]);
        }
    }

    if (lane == 0) { ws[WS_PM + bid] = m; ws[WS_PS + bid] = ssum; }
    float* pw = ws + WS_PW + (size_t)bid * H_;
#pragma unroll
    for (int i = 0; i < 4; ++i) {
        float4 o; o.x = w[i*4+0]; o.y = w[i*4+1]; o.z = w[i*4+2]; o.w = w[i*4+3];
        *(float4*)(pw + i * 128 + lane * 4) = o;
    }
}

// ---------------------------------------------------------------------------
// Kernel 3: merge chunk partials per (branch,b): global M, Z, weighted[h].
// grid: 64 blocks of 512 threads (one thread per h).
// ---------------------------------------------------------------------------
__global__ __launch_bounds__(512) void combine_kernel(float* __restrict__ ws)
{
    int pair = blockIdx.x;          // br*32 + b
    int h = threadIdx.x;
    const float* pm = ws + WS_PM + pair * NCH;
    const float* ps = ws + WS_PS + pair * NCH;

    float M = -INFINITY;
    for (int c = 0; c < NCH; ++c) M = fmaxf(M, pm[c]);
    float Z = 0.f;
    for (int c = 0; c < NCH; ++c) Z += ps[c] * __expf(pm[c] - M);

    float acc = 0.f;
    const float* pw = ws + WS_PW + (size_t)pair * NCH * H_;
    for (int c = 0; c < NCH; ++c)
        acc = fmaf(pw[(size_t)c * H_ + h], __expf(pm[c] - M), acc);

    ws[WS_WEIGHTED + (size_t)pair * H_ + h] = acc / Z;
    if (h == 0) { ws[WS_MZ + pair * 2] = M; ws[WS_MZ + pair * 2 + 1] = Z; }
}

// ---------------------------------------------------------------------------
// Kernel 4: normalize raw scores into softmax att (in-place in d_out).
// ---------------------------------------------------------------------------
__global__ __launch_bounds__(256) void normalize_att_kernel(
    float* __restrict__ out, const float* __restrict__ ws)
{
    int idx = blockIdx.x * 256 + threadIdx.x;
    if (idx >= 2 * B_ * S_) return;
    int br = idx / (B_ * S_);
    int r  = idx - br * (B_ * S_);
    int b  = r / S_;
    int pair = br * B_ + b;
    float M = ws[WS_MZ + pair * 2];
    float Z = ws[WS_MZ + pair * 2 + 1];
    float* att = out + 16384;
    att[idx] = __expf(att[idx] - M) / Z;
}

// ---------------------------------------------------------------------------
// Kernel 5: combined = tanh(concat(tree_w, tree_in, text_w, text_in) @ W_out)
// via V_WMMA_F32_16X16X4_F32.  grid: 2 Mtiles * 32 Ntiles = 64 waves.
// ---------------------------------------------------------------------------
__device__ __forceinline__ float fetchA(int b, int j,
    const float* __restrict__ wtree, const float* __restrict__ tree_in,
    const float* __restrict__ wtext, const float* __restrict__ text_in)
{
    if (j < 512)  return wtree[b * 512 + j];
    if (j < 1024) return tree_in[b * 512 + (j - 512)];
    if (j < 1536) return wtext[b * 512 + (j - 1024)];
    return text_in[b * 512 + (j - 1536)];
}

__global__ __launch_bounds__(32) void out_gemm_kernel(
    const float* __restrict__ tree_in, const float* __restrict__ text_in,
    const float* __restrict__ W_out, const float* __restrict__ ws,
    float* __restrict__ out)
{
    int bid = blockIdx.x;
    int mt = bid >> 5;
    int nt = bid & 31;
    int lane = threadIdx.x;
    int half = lane >> 4;
    int l15  = lane & 15;
    int arow = mt * 16 + l15;
    int bcol = nt * 16 + l15;

    const float* wtree = ws + WS_WEIGHTED;
    const float* wtext = ws + WS_WEIGHTED + B_ * H_;

    v8f acc = {0.f, 0.f, 0.f, 0.f, 0.f, 0.f, 0.f, 0.f};
    for (int k0 = 0; k0 < 4 * H_; k0 += 4) {
        int ka = k0 + half * 2;
        v2f a;
        a.x = fetchA(arow, ka,     wtree, tree_in, wtext, text_in);
        a.y = fetchA(arow, ka + 1, wtree, tree_in, wtext, text_in);
        v2f b;
        b.x = W_out[ka * H_ + bcol];
        b.y = W_out[(ka + 1) * H_ + bcol];
        acc = __builtin_amdgcn_wmma_f32_16x16x4_f32(false, a, false, b,
                                                    (short)0, acc, false, false);
    }
#pragma unroll
    for (int i = 0; i < 8; ++i) {
        int row = mt * 16 + i + half * 8;
        out[row * H_ + bcol] = tanhf(acc[i]);
    }
}

// ---------------------------------------------------------------------------
extern "C" void kernel_launch(void* const* d_in, const int* in_sizes, int n_in,
                              void* d_out, int out_size, void* d_ws, size_t ws_size,
                              hipStream_t stream)
{
    const float* tree_in   = (const float*)d_in[0];
    const float* tree_ctx  = (const float*)d_in[1];
    const float* text_in   = (const float*)d_in[2];
    const float* text_ctx  = (const float*)d_in[3];
    const float* tree_mask = (const float*)d_in[4];
    const float* text_mask = (const float*)d_in[5];
    const float* W_in      = (const float*)d_in[6];
    const float* W_out     = (const float*)d_in[7];
    float* out = (float*)d_out;
    float* ws  = (float*)d_ws;

    target_gemm_kernel<<<128, 32, 0, stream>>>(tree_in, text_in, W_in, ws);
    attn_partial_kernel<<<2048, 32, 0, stream>>>(tree_ctx, text_ctx,
                                                 tree_mask, text_mask, ws, out);
    combine_kernel<<<64, 512, 0, stream>>>(ws);
    normalize_att_kernel<<<(2 * B_ * S_ + 255) / 256, 256, 0, stream>>>(out, ws);
    out_gemm_kernel<<<64, 32, 0, stream>>>(tree_in, text_in, W_out, ws, out);
}